// FullModel_51805895524677
// MI455X (gfx1250) — compile-verified
//
#include <hip/hip_runtime.h>
#include <math.h>

typedef float v2f __attribute__((ext_vector_type(2)));
typedef float v8f __attribute__((ext_vector_type(8)));

#define BATCH 16

static __device__ __forceinline__ v8f wmma4(v2f a, v2f b, v8f c) {
  // V_WMMA_F32_16X16X4_F32 : D(16x16,f32) = A(16x4,f32) * B(4x16,f32) + C
  return __builtin_amdgcn_wmma_f32_16x16x4_f32(false, a, false, b, (short)0, c, false, false);
}

// ---------------------------------------------------------------- TSCM stages

__global__ void taylor_kernel(const float* __restrict__ x, const float* __restrict__ c,
                              float* __restrict__ t, int n) {
  int i = blockIdx.x * blockDim.x + threadIdx.x;
  if (i >= n) return;
  float v = x[i];
  float c0 = c[0], c1 = c[1], c2 = c[2] * 0.5f, c3 = c[3] * (1.0f / 6.0f), c4 = c[4] * (1.0f / 24.0f);
  t[i] = c0 + v * (c1 + v * (c2 + v * (c3 + v * c4)));
}

__global__ void gap_kernel(const float* __restrict__ t, float* __restrict__ gap) {
  __shared__ float sh[256];
  int plane = blockIdx.x;                    // b*5+c, 80 planes
  const float* p = t + (size_t)plane * 65536;
  float s = 0.f;
  for (int i = threadIdx.x; i < 65536; i += 256) s += p[i];
  sh[threadIdx.x] = s; __syncthreads();
  for (int k = 128; k > 0; k >>= 1) { if (threadIdx.x < k) sh[threadIdx.x] += sh[threadIdx.x + k]; __syncthreads(); }
  if (threadIdx.x == 0) gap[plane] = sh[0] * (1.0f / 65536.0f);
}

__global__ void ca_kernel(const float* __restrict__ gap,
                          const float* __restrict__ w1, const float* __restrict__ b1,
                          const float* __restrict__ w2, const float* __restrict__ b2,
                          float* __restrict__ chw) {
  int b = threadIdx.x;
  if (b >= BATCH) return;
  float g[5], h[4];
  for (int i = 0; i < 5; ++i) g[i] = gap[b * 5 + i];
  for (int j = 0; j < 4; ++j) {
    float s = b1[j];
    for (int i = 0; i < 5; ++i) s += g[i] * w1[j * 5 + i];
    h[j] = fmaxf(s, 0.f);
  }
  for (int c = 0; c < 5; ++c) {
    float s = b2[c];
    for (int j = 0; j < 4; ++j) s += h[j] * w2[c * 4 + j];
    chw[b * 5 + c] = 1.0f / (1.0f + expf(-s));
  }
}

__global__ void scale_sp_kernel(float* __restrict__ t, const float* __restrict__ chw,
                                float* __restrict__ sp) {
  int i = blockIdx.x * blockDim.x + threadIdx.x;   // 16*65536
  if (i >= BATCH * 65536) return;
  int b = i >> 16, hw = i & 65535;
  float sum = 0.f, mx = -3.4e38f;
  for (int c = 0; c < 5; ++c) {
    size_t a = ((size_t)(b * 5 + c)) * 65536 + hw;
    float v = t[a] * chw[b * 5 + c];
    t[a] = v;
    sum += v; mx = fmaxf(mx, v);
  }
  sp[((size_t)(b * 2 + 0)) * 65536 + hw] = sum * (1.0f / 5.0f);
  sp[((size_t)(b * 2 + 1)) * 65536 + hw] = mx;
}

__global__ void sa_kernel(float* __restrict__ t, const float* __restrict__ sp,
                          const float* __restrict__ w, const float* __restrict__ bias) {
  int i = blockIdx.x * blockDim.x + threadIdx.x;   // 16*65536
  if (i >= BATCH * 65536) return;
  int b = i >> 16, hw = i & 65535;
  int y = hw >> 8, x = hw & 255;
  float acc = bias[0];
  for (int r = 0; r < 7; ++r) {
    int iy = y - 3 + r; if (iy < 0 || iy >= 256) continue;
    for (int c = 0; c < 7; ++c) {
      int ix = x - 3 + c; if (ix < 0 || ix >= 256) continue;
      float s0 = sp[((size_t)(b * 2 + 0)) * 65536 + iy * 256 + ix];
      float s1 = sp[((size_t)(b * 2 + 1)) * 65536 + iy * 256 + ix];
      acc += s0 * w[r * 7 + c] + s1 * w[49 + r * 7 + c];
    }
  }
  float s = 1.0f / (1.0f + expf(-acc));
  for (int c = 0; c < 5; ++c) t[((size_t)(b * 5 + c)) * 65536 + hw] *= s;
}

__global__ void mix_kernel(const float* __restrict__ t, float* __restrict__ o,
                           const float* __restrict__ mw, const float* __restrict__ mb,
                           const float* __restrict__ g, const float* __restrict__ beta,
                           const float* __restrict__ m, const float* __restrict__ var) {
  int i = blockIdx.x * blockDim.x + threadIdx.x;   // 16*65536
  if (i >= BATCH * 65536) return;
  int b = i >> 16, hw = i & 65535;
  float in[5];
  for (int c = 0; c < 5; ++c) in[c] = t[((size_t)(b * 5 + c)) * 65536 + hw];
  for (int c = 0; c < 3; ++c) {
    float s = mb[c];
    for (int k = 0; k < 5; ++k) s += in[k] * mw[c * 5 + k];
    s = (s - m[c]) * (g[c] / sqrtf(var[c] + 1e-5f)) + beta[c];
    s = (s >= 0.f) ? s : 0.1f * s;                 // LeakyReLU(0.1)
    o[((size_t)(b * 3 + c)) * 65536 + hw] = s;
  }
}

__global__ void minmax_kernel(const float* __restrict__ t, float* __restrict__ mn, float* __restrict__ mx) {
  __shared__ float smn[256], smx[256];
  int plane = blockIdx.x;                          // 48 planes
  const float* p = t + (size_t)plane * 65536;
  float lo = 3.4e38f, hi = -3.4e38f;
  for (int i = threadIdx.x; i < 65536; i += 256) { float v = p[i]; lo = fminf(lo, v); hi = fmaxf(hi, v); }
  smn[threadIdx.x] = lo; smx[threadIdx.x] = hi; __syncthreads();
  for (int k = 128; k > 0; k >>= 1) {
    if (threadIdx.x < k) {
      smn[threadIdx.x] = fminf(smn[threadIdx.x], smn[threadIdx.x + k]);
      smx[threadIdx.x] = fmaxf(smx[threadIdx.x], smx[threadIdx.x + k]);
    }
    __syncthreads();
  }
  if (threadIdx.x == 0) { mn[plane] = smn[0]; mx[plane] = smx[0]; }
}

__global__ void cnormalize_kernel(float* __restrict__ t, const float* __restrict__ mn,
                                  const float* __restrict__ mx, int n) {
  int i = blockIdx.x * blockDim.x + threadIdx.x;
  if (i >= n) return;
  int plane = i >> 16;
  t[i] = (t[i] - mn[plane]) / (mx[plane] - mn[plane] + 1e-4f);
}

// ------------------------------------------------- generic WMMA implicit GEMM conv

template <int KH, int KW, int S, int TRANS>
__device__ __forceinline__ float im2col_val(const float* __restrict__ in, int bb, int inC,
                                            int H, int W, int oy, int ox, int pad, int k,
                                            bool pixValid, int prerelu) {
  const int KHW = KH * KW;
  int ic = k / KHW;
  int rc = k - ic * KHW;
  int r = rc / KW;
  int c = rc - r * KW;
  int iy, ix;
  bool ok = pixValid;
  if (TRANS) {
    int yn = oy + pad - r, xn = ox + pad - c;
    iy = yn / S; ix = xn / S;
    ok = ok && yn >= 0 && xn >= 0 && (yn % S) == 0 && (xn % S) == 0 && iy < H && ix < W;
  } else {
    iy = oy * S - pad + r; ix = ox * S - pad + c;
    ok = ok && iy >= 0 && iy < H && ix >= 0 && ix < W;
  }
  float v = ok ? in[((size_t)(bb * inC + ic) * H + iy) * W + ix] : 0.0f;
  if (prerelu) v = fmaxf(v, 0.0f);
  return v;
}

// Per wave: 32 outC x 16 pixels (two 16x16 WMMA tiles sharing the gathered B fragment).
// K = inC*KH*KW stepped by 4; all layers here have K % 16 == 0.
template <int KH, int KW, int S, int TRANS>
__global__ void conv2d_wmma(const float* __restrict__ in, const float* __restrict__ wt,
                            const float* __restrict__ bias, const float* __restrict__ residual,
                            float* __restrict__ out,
                            int inC, int H, int W, int outC, int outH, int outW,
                            int pad, int prerelu, int actrelu) {
  const int lane = threadIdx.x & 31;
  const int wave = (blockIdx.x * blockDim.x + threadIdx.x) >> 5;
  const int Npix = BATCH * outH * outW;
  const int nTM2 = (outC + 31) >> 5;   // pairs of 16-row M tiles
  const int nTN = (Npix + 15) >> 4;
  if (wave >= nTM2 * nTN) return;
  const int mt = wave % nTM2, nt = wave / nTM2;
  const int half = lane >> 4, l16 = lane & 15;

  // B (and C/D) column: pixel for this lane
  const int pix = nt * 16 + l16;
  const bool pixValid = pix < Npix;
  const int pp = pixValid ? pix : 0;
  const int ox = pp % outW;
  const int ty = pp / outW;
  const int oy = ty % outH;
  const int bb = ty / outH;

  // A rows: two output channels per lane (tile 0 and tile 1)
  const int m0 = mt * 32 + l16;
  const int m1 = m0 + 16;
  const bool m0v = m0 < outC;
  const bool m1v = m1 < outC;

  const int KHW = KH * KW;
  const int K = inC * KHW;

  v8f acc0, acc1;
  for (int r = 0; r < 8; ++r) { acc0[r] = 0.0f; acc1[r] = 0.0f; }

#pragma unroll 4
  for (int k0 = 0; k0 < K; k0 += 4) {
    const int ka = k0 + (half << 1);   // VGPR0 holds K=ka, VGPR1 holds K=ka+1
    v2f a0, a1, b;
    if (!TRANS) {
      // OIHW: contiguous K per output channel
      const float* wr0 = wt + (size_t)m0 * K + ka;
      const float* wr1 = wt + (size_t)m1 * K + ka;
      a0.x = m0v ? wr0[0] : 0.0f;
      a0.y = m0v ? wr0[1] : 0.0f;
      a1.x = m1v ? wr1[0] : 0.0f;
      a1.y = m1v ? wr1[1] : 0.0f;
    } else {
      // ConvTranspose weights: (in, out, kh, kw)
      int ic0 = ka / KHW, rc0 = ka - ic0 * KHW;
      int k1 = ka + 1;
      int ic1 = k1 / KHW, rc1 = k1 - ic1 * KHW;
      size_t o0 = (size_t)ic0 * outC * KHW + rc0;
      size_t o1 = (size_t)ic1 * outC * KHW + rc1;
      a0.x = m0v ? wt[o0 + (size_t)m0 * KHW] : 0.0f;
      a0.y = m0v ? wt[o1 + (size_t)m0 * KHW] : 0.0f;
      a1.x = m1v ? wt[o0 + (size_t)m1 * KHW] : 0.0f;
      a1.y = m1v ? wt[o1 + (size_t)m1 * KHW] : 0.0f;
    }
    // Gathered B fragment built once, fed to both WMMAs
    b.x = im2col_val<KH, KW, S, TRANS>(in, bb, inC, H, W, oy, ox, pad, ka, pixValid, prerelu);
    b.y = im2col_val<KH, KW, S, TRANS>(in, bb, inC, H, W, oy, ox, pad, ka + 1, pixValid, prerelu);
    acc0 = wmma4(a0, b, acc0);
    acc1 = wmma4(a1, b, acc1);
  }

  if (pixValid) {
    for (int tile = 0; tile < 2; ++tile) {
      const v8f& acc = tile ? acc1 : acc0;
      for (int r = 0; r < 8; ++r) {
        int oc = mt * 32 + tile * 16 + r + (half << 3);  // C/D: VGPR r -> M=r / r+8
        if (oc < outC) {
          float v = acc[r] + bias[oc];
          size_t addr = ((size_t)(bb * outC + oc) * outH + oy) * outW + ox;
          if (residual) v += residual[addr];
          if (actrelu) v = fmaxf(v, 0.0f);
          out[addr] = v;
        }
      }
    }
  }
}

// ------------------------------------------------------------------- VQ stages

__global__ void cnorm_kernel(const float* __restrict__ cb, float* __restrict__ cn) {
  int i = blockIdx.x * blockDim.x + threadIdx.x;
  if (i >= 512) return;
  const float* r = cb + (size_t)i * 64;
  float s = 0.f;
  for (int d = 0; d < 64; ++d) s += r[d] * r[d];
  cn[i] = s;
}

__global__ void vq_init_kernel(int* __restrict__ counts) {
  int i = blockIdx.x * blockDim.x + threadIdx.x;
  if (i < 512) counts[i] = 0;
}

// argmin_k ||z - c_k||^2 == argmin_k (||c_k||^2 - 2 z.c_k); 16 pixels x 16 codes per wave.
__global__ void vq_argmin_kernel(const float* __restrict__ zenc, const float* __restrict__ cb,
                                 const float* __restrict__ cnorm, int* __restrict__ idxOut) {
  const int lane = threadIdx.x & 31;
  const int wave = (blockIdx.x * blockDim.x + threadIdx.x) >> 5;  // 4096 waves
  const int half = lane >> 4, l16 = lane & 15;
  const int pix = wave * 16 + l16;                                 // < 65536
  const int bb = pix >> 12, hw = pix & 4095;

  float best = 3.4e38f;
  int bestIdx = 0;
  for (int ct = 0; ct < 32; ++ct) {
    const int c0 = ct << 4;
    v8f acc;
    for (int r = 0; r < 8; ++r) acc[r] = 0.f;
#pragma unroll
    for (int k0 = 0; k0 < 64; k0 += 4) {
      const int ka = k0 + (half << 1);
      v2f a, b;
      const float* cr = cb + (size_t)(c0 + l16) * 64 + ka;
      a.x = cr[0]; a.y = cr[1];
      size_t za = ((size_t)(bb * 64 + ka)) * 4096 + hw;   // zenc NCHW, channel = dim
      b.x = zenc[za];
      b.y = zenc[za + 4096];
      acc = wmma4(a, b, acc);
    }
    for (int r = 0; r < 8; ++r) {
      int code = c0 + r + (half << 3);
      float d = cnorm[code] - 2.0f * acc[r];
      if (d < best) { best = d; bestIdx = code; }   // ascending order -> first-min semantics
    }
  }
  float ob = __shfl_xor(best, 16, 32);
  int oi = __shfl_xor(bestIdx, 16, 32);
  if (ob < best || (ob == best && oi < bestIdx)) { best = ob; bestIdx = oi; }
  if (half == 0) idxOut[pix] = bestIdx;
}

__global__ void vq_stats_kernel(const float* __restrict__ zenc, const float* __restrict__ cb,
                                const int* __restrict__ idx, float* __restrict__ zq,
                                int* __restrict__ counts, float* __restrict__ partials) {
  __shared__ float sh[256];
  int p = blockIdx.x * 256 + threadIdx.x;   // < 65536
  int bb = p >> 12, hw = p & 4095;
  int code = idx[p];
  const float* cw = cb + (size_t)code * 64;
  float s = 0.f;
  for (int d = 0; d < 64; ++d) {
    size_t a = ((size_t)(bb * 64 + d)) * 4096 + hw;
    float q = cw[d];
    float z = zenc[a];
    zq[a] = q;                               // straight-through: zq == q numerically
    float df = z - q;
    s += df * df;
  }
  atomicAdd(&counts[code], 1);               // integer: deterministic
  sh[threadIdx.x] = s; __syncthreads();
  for (int k = 128; k > 0; k >>= 1) { if (threadIdx.x < k) sh[threadIdx.x] += sh[threadIdx.x + k]; __syncthreads(); }
  if (threadIdx.x == 0) partials[blockIdx.x] = sh[0];
}

__global__ void vq_finalize_kernel(const int* __restrict__ counts, const float* __restrict__ partials,
                                   float* __restrict__ out2) {
  __shared__ float sh[512];
  __shared__ float sp[256];
  int t = threadIdx.x;   // 512 threads
  float pr = (float)counts[t] * (1.0f / 65536.0f);
  sh[t] = pr * logf(pr + 1e-10f);
  if (t < 256) sp[t] = partials[t];
  __syncthreads();
  for (int k = 256; k > 0; k >>= 1) { if (t < k) sh[t] += sh[t + k]; __syncthreads(); }
  for (int k = 128; k > 0; k >>= 1) { if (t < k) sp[t] += sp[t + k]; __syncthreads(); }
  if (t == 0) {
    out2[0] = 0.25f * sp[0] * (1.0f / (65536.0f * 64.0f));  // COMMIT * mean((z-q)^2)
    out2[1] = expf(-sh[0]);                                 // perplexity
  }
}

// ---------------------------------------------------------------------- driver

static inline int cdiv(int a, int b) { return (a + b - 1) / b; }

#define LAUNCH_CONV(KH, KW, S, TRANS, in_, w_, b_, res_, out_, inC_, H_, W_, outC_, outH_, outW_, pad_, prerelu_, act_) \
  do {                                                                                                                  \
    int Npix_ = BATCH * (outH_) * (outW_);                                                                              \
    int tiles_ = cdiv((outC_), 32) * cdiv(Npix_, 16);                                                                   \
    conv2d_wmma<KH, KW, S, TRANS><<<cdiv(tiles_, 8), 256, 0, stream>>>(                                                 \
        in_, w_, b_, res_, out_, inC_, H_, W_, outC_, outH_, outW_, pad_, prerelu_, act_);                              \
  } while (0)

extern "C" void kernel_launch(void* const* d_in, const int* in_sizes, int n_in,
                              void* d_out, int out_size, void* d_ws, size_t ws_size,
                              hipStream_t stream) {
  (void)in_sizes; (void)n_in; (void)out_size; (void)ws_size;
  const float* x        = (const float*)d_in[0];
  const float* taylor_c = (const float*)d_in[1];
  const float* ca_w1 = (const float*)d_in[2];
  const float* ca_b1 = (const float*)d_in[3];
  const float* ca_w2 = (const float*)d_in[4];
  const float* ca_b2 = (const float*)d_in[5];
  const float* sa_w  = (const float*)d_in[6];
  const float* sa_b  = (const float*)d_in[7];
  const float* mix_w = (const float*)d_in[8];
  const float* mix_b = (const float*)d_in[9];
  const float* bn_g  = (const float*)d_in[10];
  const float* bn_b  = (const float*)d_in[11];
  const float* bn_m  = (const float*)d_in[12];
  const float* bn_v  = (const float*)d_in[13];
  const float* e1_w = (const float*)d_in[14];  const float* e1_b = (const float*)d_in[15];
  const float* e2_w = (const float*)d_in[16];  const float* e2_b = (const float*)d_in[17];
  const float* e3_w = (const float*)d_in[18];  const float* e3_b = (const float*)d_in[19];
  const float* er1a_w = (const float*)d_in[20]; const float* er1a_b = (const float*)d_in[21];
  const float* er1b_w = (const float*)d_in[22]; const float* er1b_b = (const float*)d_in[23];
  const float* er2a_w = (const float*)d_in[24]; const float* er2a_b = (const float*)d_in[25];
  const float* er2b_w = (const float*)d_in[26]; const float* er2b_b = (const float*)d_in[27];
  const float* ep_w = (const float*)d_in[28];  const float* ep_b = (const float*)d_in[29];
  const float* codebook = (const float*)d_in[30];
  const float* d1_w = (const float*)d_in[31];  const float* d1_b = (const float*)d_in[32];
  const float* dr1a_w = (const float*)d_in[33]; const float* dr1a_b = (const float*)d_in[34];
  const float* dr1b_w = (const float*)d_in[35]; const float* dr1b_b = (const float*)d_in[36];
  const float* dr2a_w = (const float*)d_in[37]; const float* dr2a_b = (const float*)d_in[38];
  const float* dr2b_w = (const float*)d_in[39]; const float* dr2b_b = (const float*)d_in[40];
  const float* d2_w = (const float*)d_in[41];  const float* d2_b = (const float*)d_in[42];
  const float* d3_w = (const float*)d_in[43];  const float* d3_b = (const float*)d_in[44];

  float* out = (float*)d_out;   // [recon (5,242,880) | vq_loss | perplexity]

  // workspace arena (floats), lifetimes allow reuse; total ~176.4 MB
  float* ws = (float*)d_ws;
  float* B0 = ws;                 // 16,777,216 : t | h1 | enc-res ping | decoder h (pre-d3)
  float* B1 = ws + 16777216;      //  8,388,608 : t3 | h2 | dec-res pong
  float* B2 = ws + 25165824;      //  8,388,608 : h3/x | x_res | d1 out | dec-res ping
  float* Bz = ws + 33554432;      //  4,194,304 : zenc
  float* Bq = ws + 37748736;      //  4,194,304 : zq
  float* B6 = ws + 41943040;      //  2,097,152 : sp_in | res-block hidden (32ch)
  float* SM = ws + 44040192;      // small region
  float* gapB = SM;               // 80
  float* chwB = SM + 80;          // 80
  float* mnB  = SM + 160;         // 48
  float* mxB  = SM + 208;         // 48
  int*   countsB = (int*)(SM + 256);   // 512 ints
  float* partB  = SM + 768;       // 256
  float* cnormB = SM + 1024;      // 512
  int*   idxB   = (int*)(SM + 1536);   // 65536 ints

  const int NP = BATCH * 65536;   // pixels at 256x256

  // ---- TSCM ----
  taylor_kernel<<<cdiv(5242880, 256), 256, 0, stream>>>(x, taylor_c, B0, 5242880);
  gap_kernel<<<80, 256, 0, stream>>>(B0, gapB);
  ca_kernel<<<1, 32, 0, stream>>>(gapB, ca_w1, ca_b1, ca_w2, ca_b2, chwB);
  scale_sp_kernel<<<cdiv(NP, 256), 256, 0, stream>>>(B0, chwB, B6);
  sa_kernel<<<cdiv(NP, 256), 256, 0, stream>>>(B0, B6, sa_w, sa_b);
  mix_kernel<<<cdiv(NP, 256), 256, 0, stream>>>(B0, B1, mix_w, mix_b, bn_g, bn_b, bn_m, bn_v);
  minmax_kernel<<<48, 256, 0, stream>>>(B1, mnB, mxB);
  cnormalize_kernel<<<cdiv(48 * 65536, 256), 256, 0, stream>>>(B1, mnB, mxB, 48 * 65536);

  // ---- Encoder ----
  LAUNCH_CONV(4, 4, 2, 0, B1, e1_w, e1_b, nullptr, B0, 3, 256, 256, 64, 128, 128, 1, 0, 1);
  LAUNCH_CONV(4, 4, 2, 0, B0, e2_w, e2_b, nullptr, B1, 64, 128, 128, 128, 64, 64, 1, 0, 1);
  LAUNCH_CONV(3, 3, 1, 0, B1, e3_w, e3_b, nullptr, B2, 128, 64, 64, 128, 64, 64, 1, 0, 0);
  // res block 1: x(B2) -> B0
  LAUNCH_CONV(3, 3, 1, 0, B2, er1a_w, er1a_b, nullptr, B6, 128, 64, 64, 32, 64, 64, 1, 1, 1);
  LAUNCH_CONV(1, 1, 1, 0, B6, er1b_w, er1b_b, B2, B0, 32, 64, 64, 128, 64, 64, 0, 0, 0);
  // res block 2: x(B0) -> B2
  LAUNCH_CONV(3, 3, 1, 0, B0, er2a_w, er2a_b, nullptr, B6, 128, 64, 64, 32, 64, 64, 1, 1, 1);
  LAUNCH_CONV(1, 1, 1, 0, B6, er2b_w, er2b_b, B0, B2, 32, 64, 64, 128, 64, 64, 0, 0, 0);
  // stack-final relu fused as prerelu of ep
  LAUNCH_CONV(1, 1, 1, 0, B2, ep_w, ep_b, nullptr, Bz, 128, 64, 64, 64, 64, 64, 0, 1, 0);

  // ---- VQ ----
  cnorm_kernel<<<2, 256, 0, stream>>>(codebook, cnormB);
  vq_init_kernel<<<2, 256, 0, stream>>>(countsB);
  vq_argmin_kernel<<<512, 256, 0, stream>>>(Bz, codebook, cnormB, idxB);
  vq_stats_kernel<<<256, 256, 0, stream>>>(Bz, codebook, idxB, Bq, countsB, partB);
  vq_finalize_kernel<<<1, 512, 0, stream>>>(countsB, partB, out + 5242880);

  // ---- Decoder ----
  LAUNCH_CONV(3, 3, 1, 1, Bq, d1_w, d1_b, nullptr, B2, 64, 64, 64, 128, 64, 64, 1, 0, 0);
  // res block 1: x(B2) -> B1
  LAUNCH_CONV(3, 3, 1, 0, B2, dr1a_w, dr1a_b, nullptr, B6, 128, 64, 64, 32, 64, 64, 1, 1, 1);
  LAUNCH_CONV(1, 1, 1, 0, B6, dr1b_w, dr1b_b, B2, B1, 32, 64, 64, 128, 64, 64, 0, 0, 0);
  // res block 2: x(B1) -> B2
  LAUNCH_CONV(3, 3, 1, 0, B1, dr2a_w, dr2a_b, nullptr, B6, 128, 64, 64, 32, 64, 64, 1, 1, 1);
  LAUNCH_CONV(1, 1, 1, 0, B6, dr2b_w, dr2b_b, B1, B2, 32, 64, 64, 128, 64, 64, 0, 0, 0);
  // d2: stack-final relu fused as prerelu; own relu after
  LAUNCH_CONV(4, 4, 2, 1, B2, d2_w, d2_b, nullptr, B0, 128, 64, 64, 64, 128, 128, 1, 1, 1);
  // d3: recon straight to d_out
  LAUNCH_CONV(4, 4, 2, 1, B0, d3_w, d3_b, nullptr, out, 64, 128, 128, 5, 256, 256, 1, 0, 0);
}